// MoEGate_89773406421362
// MI455X (gfx1250) — compile-verified
//
#include <hip/hip_runtime.h>

typedef float v2f __attribute__((ext_vector_type(2)));
typedef float v8f __attribute__((ext_vector_type(8)));
typedef int   v4i_t __attribute__((vector_size(4 * sizeof(int))));
typedef int   v2i_t __attribute__((vector_size(2 * sizeof(int))));

#define T_TOKENS 16384
#define HDIM     2048
#define NEXP     64
#define SEQ      4096
#define NBATCH   4
#define TOPK     6
#define BM       128     // tokens per workgroup
#define BK       32      // K chunk staged in LDS (double buffered)
#define NCHUNK   (HDIM / BK)
#define ASTRIDE  36      // 36*4 = 144 B row stride: 16B aligned, bank-spread
#define BSTRIDE  130     // k-pair row stride (floats); Bs[(k/2)*130 + n*2 + (k&1)]
#define LSTRIDE  65

#ifndef __has_builtin
#define __has_builtin(x) 0
#endif
#if __has_builtin(__builtin_amdgcn_global_load_async_to_lds_b128) && \
    __has_builtin(__builtin_amdgcn_global_load_async_to_lds_b64)
#define HAVE_ASYNC_LDS 1
#else
#define HAVE_ASYNC_LDS 0
#endif

// generic -> AS1 / AS3 typed-vector pointers via integer casts
// (generic global addr == global vaddr; low 32 bits of a generic LDS address
//  are the LDS byte offset, ISA 10.2)
#define AS1_V4I(p) ((__attribute__((address_space(1))) v4i_t*)(unsigned long long)(p))
#define AS3_V4I(p) ((__attribute__((address_space(3))) v4i_t*)(unsigned int)(unsigned long long)(p))
#define AS1_V2I(p) ((__attribute__((address_space(1))) v2i_t*)(unsigned long long)(p))
#define AS3_V2I(p) ((__attribute__((address_space(3))) v2i_t*)(unsigned int)(unsigned long long)(p))

union GateSmem {
    struct {
        float As[2][BM * ASTRIDE];           // A tiles [buf][row][k]
        float Bs[2][(BK / 2) * BSTRIDE];     // B tiles [buf][k/2][n*2 + (k&1)]
    } g;
    float Ls[BM * LSTRIDE];                  // logits [row][e], reuses GEMM space
};

__device__ __forceinline__ void wait_async_lds() {
#if HAVE_ASYNC_LDS
#if __has_builtin(__builtin_amdgcn_s_wait_asynccnt)
    __builtin_amdgcn_s_wait_asynccnt(0);
#else
    asm volatile("s_wait_asynccnt 0x0" ::: "memory");
#endif
#endif
}

__device__ __forceinline__ void stage_tiles(float* __restrict__ As, float* __restrict__ Bs,
                                            const float* __restrict__ x,
                                            const float* __restrict__ w,
                                            int wgTok, int kk, int tid) {
#if HAVE_ASYNC_LDS
    // A tile: 128 rows x 32 floats = 1024 float4 slots -> 4 async b128 per thread
    #pragma unroll
    for (int it = 0; it < 4; ++it) {
        int idx = it * 256 + tid;
        int row = idx >> 3;
        int c4  = idx & 7;
        const float* gp = &x[(size_t)(wgTok + row) * HDIM + kk + c4 * 4];
        float*       lp = &As[row * ASTRIDE + c4 * 4];
        __builtin_amdgcn_global_load_async_to_lds_b128(AS1_V4I(gp), AS3_V4I(lp), 0, 0);
    }
    // B tile transposed by k-pairs: 64 experts x 16 pairs -> 4 async b64 per thread
    #pragma unroll
    for (int it = 0; it < 4; ++it) {
        int idx = it * 256 + tid;
        int n   = idx >> 4;          // expert 0..63
        int kp  = idx & 15;          // k-pair 0..15
        const float* gp = &w[(size_t)n * HDIM + kk + kp * 2];
        float*       lp = &Bs[kp * BSTRIDE + n * 2];
        __builtin_amdgcn_global_load_async_to_lds_b64(AS1_V2I(gp), AS3_V2I(lp), 0, 0);
    }
#else
    // synchronous fallback: VGPR staging
    #pragma unroll
    for (int it = 0; it < 4; ++it) {
        int idx = it * 256 + tid;
        int row = idx >> 3;
        int c4  = idx & 7;
        const float4 v = *(const float4*)&x[(size_t)(wgTok + row) * HDIM + kk + c4 * 4];
        *(float4*)&As[row * ASTRIDE + c4 * 4] = v;
    }
    #pragma unroll
    for (int it = 0; it < 2; ++it) {
        int idx = it * 256 + tid;
        int n   = idx >> 3;          // expert 0..63
        int c4  = idx & 7;           // float4 chunk 0..7
        const float4 v = *(const float4*)&w[(size_t)n * HDIM + kk + c4 * 4];
        v2f lo = {v.x, v.y}, hi = {v.z, v.w};
        *(v2f*)&Bs[(c4 * 2 + 0) * BSTRIDE + n * 2] = lo;
        *(v2f*)&Bs[(c4 * 2 + 1) * BSTRIDE + n * 2] = hi;
    }
    __builtin_prefetch(&x[(size_t)(wgTok + (tid >> 1)) * HDIM + kk + (tid & 1) * 16], 0, 1);
#endif
}

__device__ __forceinline__ void compute_chunk(const float* __restrict__ As,
                                              const float* __restrict__ Bs,
                                              v8f acc[4], int mBase, int mRow, int kHalf) {
    #pragma unroll
    for (int ks = 0; ks < BK; ks += 4) {
        const int ka = ks + kHalf;
        // A frag: lanes0-15 -> K=ks,ks+1 ; lanes16-31 -> K=ks+2,ks+3 (ISA 7.12.2)
        v2f a = *(const v2f*)&As[(mBase + mRow) * ASTRIDE + ka];
        #pragma unroll
        for (int nt = 0; nt < 4; ++nt) {
            // single aligned ds_load_b64: (k, k+1) pair is contiguous in LDS
            v2f b = *(const v2f*)&Bs[(ka >> 1) * BSTRIDE + (nt * 16 + mRow) * 2];
            acc[nt] = __builtin_amdgcn_wmma_f32_16x16x4_f32(
                /*neg_a=*/false, a, /*neg_b=*/false, b,
                /*c_mod=*/(short)0, acc[nt], /*reuse_a=*/false, /*reuse_b=*/false);
        }
    }
}

__global__ __launch_bounds__(256) void moe_gate_wmma_kernel(
    const float* __restrict__ x,      // [T, H]
    const float* __restrict__ w,      // [E, H]
    int*   __restrict__ out_idx,      // [T, 6]
    float* __restrict__ out_wt,       // [T, 6]
    float* __restrict__ ws)           // [2][4][64]: score sums, counts
{
    __shared__ GateSmem sm;
    __shared__ float ssum[NEXP];
    __shared__ float scnt[NEXP];

    const int tid   = threadIdx.x;
    const int lane  = tid & 31;
    const int wave  = tid >> 5;
    const int wgTok = blockIdx.x * BM;

    if (tid < NEXP) { ssum[tid] = 0.0f; scnt[tid] = 0.0f; }

    const int mBase = wave * 16;        // each wave owns one 16-row M tile
    const int mRow  = lane & 15;        // row (A) / col (B,C) within tile
    const int kHalf = (lane >> 4) * 2;  // lanes 16-31 hold K+2,K+3

    v8f acc[4] = {};                    // 16x16 f32 C tiles for N = 0,16,32,48

    // ---- software-pipelined, double-buffered K loop ----
    stage_tiles(sm.g.As[0], sm.g.Bs[0], x, w, wgTok, 0, tid);
    for (int c = 0; c < NCHUNK; ++c) {
        wait_async_lds();               // my wave's async tile writes done
        __syncthreads();                // everyone's writes visible, prev compute done
        if (c + 1 < NCHUNK)
            stage_tiles(sm.g.As[(c + 1) & 1], sm.g.Bs[(c + 1) & 1],
                        x, w, wgTok, (c + 1) * BK, tid);
        compute_chunk(sm.g.As[c & 1], sm.g.Bs[c & 1], acc, mBase, mRow, kHalf);
    }
    __syncthreads();

    // ---- spill logits to LDS: C layout -> row = mBase + r (+8 for hi lanes) ----
    #pragma unroll
    for (int nt = 0; nt < 4; ++nt) {
        #pragma unroll
        for (int r = 0; r < 8; ++r) {
            int row = mBase + r + (lane >> 4) * 8;
            sm.Ls[row * LSTRIDE + nt * 16 + mRow] = acc[nt][r];
        }
    }
    __syncthreads();

    // ---- per-token softmax + top-6 + renorm + aux accumulation ----
    if (tid < BM) {
        float* L = &sm.Ls[tid * LSTRIDE];
        float mx = -1e30f;
        #pragma unroll 8
        for (int e = 0; e < NEXP; ++e) mx = fmaxf(mx, L[e]);
        float s = 0.0f;
        #pragma unroll 8
        for (int e = 0; e < NEXP; ++e) s += __expf(L[e] - mx);
        const float inv = 1.0f / s;
        #pragma unroll 8
        for (int e = 0; e < NEXP; ++e) {
            float p = __expf(L[e] - mx) * inv;   // softmax score
            L[e] = p;
            atomicAdd(&ssum[e], p);              // for mean_scores
        }
        unsigned long long used = 0ull;
        float tw[TOPK]; int ti[TOPK]; float tsum = 0.0f;
        #pragma unroll
        for (int j = 0; j < TOPK; ++j) {
            float bv = -1.0f; int bi = 0;
            for (int e = 0; e < NEXP; ++e) {
                if (!((used >> e) & 1ull)) {
                    float p = L[e];
                    if (p > bv) { bv = p; bi = e; }
                }
            }
            used |= (1ull << bi);
            tw[j] = bv; ti[j] = bi; tsum += bv;
            atomicAdd(&scnt[bi], 1.0f);          // for ce counts
        }
        const float invt = 1.0f / (tsum + 1e-20f);
        const int gt = wgTok + tid;
        #pragma unroll
        for (int j = 0; j < TOPK; ++j) {
            out_idx[gt * TOPK + j] = ti[j];
            out_wt [gt * TOPK + j] = tw[j] * invt;
        }
    }
    __syncthreads();

    // ---- one global atomic per (b, e) per workgroup ----
    if (tid < NEXP) {
        const int b = wgTok / SEQ;   // whole WG is inside one batch
        atomicAdd(&ws[b * NEXP + tid],                 ssum[tid]);
        atomicAdd(&ws[NBATCH * NEXP + b * NEXP + tid], scnt[tid]);
    }
}

__global__ void moe_aux_kernel(const float* __restrict__ ws,
                               float* __restrict__ out_loss)
{
    __shared__ float red[NEXP];
    const int e = threadIdx.x;
    const float ce_scale = (float)NEXP / ((float)SEQ * (float)TOPK);
    const float ms_scale = 1.0f / (float)SEQ;
    float part = 0.0f;
    #pragma unroll
    for (int b = 0; b < NBATCH; ++b) {
        float cnt = ws[NBATCH * NEXP + b * NEXP + e];
        float sms = ws[b * NEXP + e];
        part += (cnt * ce_scale) * (sms * ms_scale);
    }
    red[e] = part;
    __syncthreads();
    if (e == 0) {
        float t = 0.0f;
        for (int i = 0; i < NEXP; ++i) t += red[i];
        *out_loss = 0.001f * (t / (float)NBATCH);
    }
}

extern "C" void kernel_launch(void* const* d_in, const int* in_sizes, int n_in,
                              void* d_out, int out_size, void* d_ws, size_t ws_size,
                              hipStream_t stream) {
    const float* x = (const float*)d_in[0];   // [4,4096,2048] fp32
    const float* w = (const float*)d_in[1];   // [64,2048] fp32

    int*   out_idx  = (int*)d_out;                               // 98304 int32
    float* out_wt   = (float*)d_out + T_TOKENS * TOPK;           // 98304 f32
    float* out_loss = (float*)d_out + 2 * T_TOKENS * TOPK;       // 1 f32
    float* ws       = (float*)d_ws;                              // 512 f32

    (void)hipMemsetAsync(ws, 0, 2 * NBATCH * NEXP * sizeof(float), stream);
    moe_gate_wmma_kernel<<<T_TOKENS / BM, 256, 0, stream>>>(x, w, out_idx, out_wt, ws);
    moe_aux_kernel<<<1, NEXP, 0, stream>>>(ws, out_loss);
}